// Mahalanobis_8375186227263
// MI455X (gfx1250) — compile-verified
//
#include <hip/hip_runtime.h>

typedef float v2f __attribute__((ext_vector_type(2)));
typedef float v8f __attribute__((ext_vector_type(8)));

#define D        64
#define CLS      10
#define BPAD     16          // class columns padded to 16
#define KF       16          // 64 / 4 k-fragments
#define NT       5           // 80 / 16 column tiles (4 for fP, 1 for cross)
#define ASTRIDE  68          // padded LDS row stride (words) -> conflict-free ds_load_b64
#define WAVES    8
#define ROWS_PER_WAVE  16
#define ROWS_PER_BLOCK 128
#define B_SWZ_FLOATS (KF * NT * 32 * 2)   // 5120 floats = 20KB, pre-swizzled B fragments
#define BIAS_OFF     B_SWZ_FLOATS

// ---------------- prep: muP, bias, and WMMA-swizzled B = [P | muP^T] ----------------
__global__ __launch_bounds__(256) void mahal_prep(const float* __restrict__ mean,
                                                  const float* __restrict__ prec,
                                                  float* __restrict__ ws) {
    __shared__ float sMuP[CLS * D];
    const int tid = threadIdx.x;

    for (int idx = tid; idx < CLS * D; idx += 256) {
        const int c = idx / D, d = idx % D;
        float s = 0.f;
        for (int k = 0; k < D; ++k) s += mean[c * D + k] * prec[k * D + d];
        sMuP[idx] = s;
    }
    __syncthreads();

    if (tid < BPAD) {
        float b;
        if (tid < CLS) {
            float q = 0.f;
            for (int d0 = 0; d0 < D; ++d0) q += sMuP[tid * D + d0] * mean[tid * D + d0];
            b = -0.5f * q;
        } else {
            b = -__builtin_inff();   // padded classes never win the max
        }
        ws[BIAS_OFF + tid] = b;
    }

    // B fragment layout (mirrors 32-bit A layout): entry ((kf*NT+t)*32 + lane)*2 + v
    // holds B[k = 4*kf + 2*(lane>=16) + v][j = 16*t + (lane&15)]
    for (int idx = tid; idx < B_SWZ_FLOATS; idx += 256) {
        const int v    = idx & 1;
        const int lane = (idx >> 1) & 31;
        const int rest = idx >> 6;
        const int t    = rest % NT;
        const int kf   = rest / NT;
        const int k    = 4 * kf + 2 * (lane >> 4) + v;
        const int j    = 16 * t + (lane & 15);
        float val;
        if (j < D)               val = prec[k * D + j];        // P columns
        else if (j - D < CLS)    val = sMuP[(j - D) * D + k];  // muP^T columns
        else                     val = 0.f;                    // pad
        ws[idx] = val;
    }
}

// ---------------- main: per-wave 16-row tile, fp32 WMMA ----------------
__global__ __launch_bounds__(256) void mahal_main(const float* __restrict__ F,
                                                  const float* __restrict__ ws,
                                                  const float* __restrict__ alpha,
                                                  float* __restrict__ out) {
    __shared__ float sB[B_SWZ_FLOATS];
    __shared__ float sBias[BPAD];
    __shared__ float sA[WAVES][ROWS_PER_WAVE * ASTRIDE];

    const int tid = threadIdx.x;

    // cooperative load of pre-swizzled B (20KB) + bias into LDS
    {
        const float4* wsv = reinterpret_cast<const float4*>(ws);
        float4*       sBv = reinterpret_cast<float4*>(sB);
        #pragma unroll
        for (int i = 0; i < 5; ++i) {
            const int e = i * 256 + tid;
            sBv[e] = wsv[e];
        }
        if (tid < BPAD) sBias[tid] = ws[BIAS_OFF + tid];
    }

    const int w    = tid >> 5;
    const int lane = tid & 31;
    const int h    = lane >> 4;   // half-wave select (K pair / row-half)
    const int j0   = lane & 15;
    const long long rowbase = ((long long)blockIdx.x * WAVES + w) * ROWS_PER_WAVE;

    // stage this wave's 16x64 fp32 tile (contiguous 1024 floats) -> LDS, coalesced b128
    {
        const float4* src4 = reinterpret_cast<const float4*>(F + rowbase * D);
        float* dstA = sA[w];
        #pragma unroll
        for (int i = 0; i < 8; ++i) {
            const int e   = i * 32 + lane;     // float4 index within tile
            const float4 vv = src4[e];
            const int row = e >> 4;
            const int col = (e & 15) * 4;
            *reinterpret_cast<float4*>(&dstA[row * ASTRIDE + col]) = vv;
        }
    }
    __syncthreads();

    const float* A = sA[w];

    // A fragments: lane(0-15)=row M, half selects K pair -> single ds_load_b64 each
    v2f afr[KF];
    #pragma unroll
    for (int kf = 0; kf < KF; ++kf)
        afr[kf] = *reinterpret_cast<const v2f*>(&A[j0 * ASTRIDE + 4 * kf + 2 * h]);

    v8f acc[NT] = {};
    #pragma unroll
    for (int t = 0; t < NT; ++t) {
        #pragma unroll
        for (int kf = 0; kf < KF; ++kf) {
            const v2f bf = *reinterpret_cast<const v2f*>(&sB[((kf * NT + t) * 32 + lane) * 2]);
            acc[t] = __builtin_amdgcn_wmma_f32_16x16x4_f32(false, afr[kf],
                                                           false, bf,
                                                           (short)0, acc[t],
                                                           false, false);
        }
    }

    // Epilogue: D layout is lane = column N, VGPR g = row (g + 8h).
    // qf[row] = sum_cols fP*f  (butterfly-sum over the 16-lane half)
    // class score = max_c (cross_c - 0.5*qmu_c)  (butterfly-max)
    const float bias = sBias[j0];
    const float al   = alpha[0];
    float score[8];
    #pragma unroll
    for (int g = 0; g < 8; ++g) {
        const int M = g + 8 * h;
        float s = 0.f;
        #pragma unroll
        for (int t = 0; t < 4; ++t)
            s += acc[t][g] * A[M * ASTRIDE + 16 * t + j0];
        #pragma unroll
        for (int m = 1; m < 16; m <<= 1) s += __shfl_xor(s, m, 32);

        float v = acc[4][g] + bias;
        #pragma unroll
        for (int m = 1; m < 16; m <<= 1) v = fmaxf(v, __shfl_xor(v, m, 32));

        score[g] = al * (-0.5f * s + v);
    }

    #pragma unroll
    for (int g = 0; g < 8; ++g)
        if (j0 == g) out[rowbase + 8 * h + g] = score[g];
}

extern "C" void kernel_launch(void* const* d_in, const int* in_sizes, int n_in,
                              void* d_out, int out_size, void* d_ws, size_t ws_size,
                              hipStream_t stream) {
    const float* F     = (const float*)d_in[0];  // [N, 64]
    const float* mean  = (const float*)d_in[1];  // [10, 64]
    const float* prec  = (const float*)d_in[2];  // [64, 64]
    const float* alpha = (const float*)d_in[3];  // [1]
    float* out = (float*)d_out;                  // [N]
    float* ws  = (float*)d_ws;                   // >= (5120+16) floats

    const int N = in_sizes[0] / D;               // 2,097,152

    mahal_prep<<<1, 256, 0, stream>>>(mean, prec, ws);
    mahal_main<<<N / ROWS_PER_BLOCK, 256, 0, stream>>>(F, ws, alpha, out);
}